// DiscoMixerBlock_26869315403880
// MI455X (gfx1250) — compile-verified
//
#include <hip/hip_runtime.h>
#include <hip/hip_bf16.h>

typedef __attribute__((ext_vector_type(2))) float v2f;
typedef __attribute__((ext_vector_type(8))) float v8f;

#define BB    2
#define HH_   90
#define WW    180
#define PP    (HH_ * WW)   /* 16200 */
#define CC    64
#define MD    66
#define SHN   16
#define KK    25
#define NNZN  25
#define HHID  32
#define FFNH  256

__device__ __forceinline__ float gelu_f(float v) {
    return 0.5f * v * (1.0f + erff(v * 0.70710678118654752440f));
}

// D = A(16x4 f32) * B(4x16 f32) + C(16x16 f32), wave32 WMMA
__device__ __forceinline__ v8f wmma_f32(v2f a, v2f b, v8f c) {
    return __builtin_amdgcn_wmma_f32_16x16x4_f32(false, a, false, b, (short)0, c,
                                                 false, false);
}

// ---------------------------------------------------------------------------
// Kernel 1: latitude modulation MLP  m = gelu(scr@W1+b1)@W2+b2  -> (B*H, 32)
// ---------------------------------------------------------------------------
__global__ __launch_bounds__(32)
void mod_kernel(const float* __restrict__ x,
                const float* __restrict__ lw1, const float* __restrict__ lb1,
                const float* __restrict__ lw2, const float* __restrict__ lb2,
                float* __restrict__ mod)
{
    const int row = blockIdx.x;          // b*H + h
    const int j   = threadIdx.x;         // 0..31
    const float* xp = x + ((long)row * WW) * MD + CC;  // sin_cos @ w=0
    const float s0 = xp[0], s1 = xp[1];
    float acc = lb2[j];
    for (int i = 0; i < 2 * SHN; ++i) {
        float t = s0 * lw1[i] + s1 * lw1[2 * SHN + i] + lb1[i];
        acc += gelu_f(t) * lw2[i * (2 * SHN) + j];
    }
    mod[row * (2 * SHN) + j] = acc;
}

// ---------------------------------------------------------------------------
// Kernel 2: per-point disco conv + FiLM + per-head MLP + residual (all WMMA)
//   Phase B: Bmat(32x16) = basis_p^T(32x28) @ disco_w^T(28x16)   : 14 wmma
//   Phase C: dout(64x16) = gather(64x28)    @ Bmat(28x16)        : 28 wmma
//   Phase E: h1(64x32)   = blockdiag(F)(64x32) @ blockdiag(W1)   : 64 wmma
// ---------------------------------------------------------------------------
__global__ __launch_bounds__(32)
void disco_head_kernel(const float* __restrict__ x, const int* __restrict__ nbr,
                       const float* __restrict__ basis,
                       const float* __restrict__ disco_w,
                       const float* __restrict__ disco_b,
                       const float* __restrict__ hw1, const float* __restrict__ hb1,
                       const float* __restrict__ hw2, const float* __restrict__ hb2,
                       const float* __restrict__ mod, float* __restrict__ xln)
{
    __shared__ int   nbrx[28];          // 25 real neighbors + 3 clamped (safe addr)
    __shared__ float bT[32 * 28];       // basis[k,p,n] as [n][k], zero padded
    __shared__ float dws[16 * 28];      // disco_w [o][k], zero padded
    __shared__ float bmat[32 * 16];     // Bmat[n][o]; rows 25..31 == 0 naturally
    __shared__ float fbuf[64 * 17];     // FiLM'd dout
    __shared__ float h1buf[64 * 33];    // gelu'd head hidden

    const int l = threadIdx.x;
    const int q = blockIdx.x;           // b*P + p
    const int b = q / PP;
    const int p = q - b * PP;

    for (int e = l; e < 32 * 28; e += 32) bT[e] = 0.f;
    for (int e = l; e < 16 * 28; e += 32) dws[e] = 0.f;
    __syncthreads();

    if (l < 28) nbrx[l] = nbr[p * NNZN + ((l < NNZN) ? l : 0)];
    for (int e = l; e < KK * KK; e += 32) {
        const int k = e / KK, n = e - k * KK;
        bT[n * 28 + k] = basis[(long)k * (PP * NNZN) + (long)p * NNZN + n];
    }
    for (int e = l; e < SHN * KK; e += 32) {
        const int o = e / KK, k = e - o * KK;
        dws[o * 28 + k] = disco_w[e];
    }
    __syncthreads();

    const bool hi = l >= 16;
    const int  lr = l & 15;

    // ---- Phase B: Bmat via WMMA -------------------------------------------
    {
        v8f zb = {0.f, 0.f, 0.f, 0.f, 0.f, 0.f, 0.f, 0.f};
        v8f accb[2] = {zb, zb};
        #pragma unroll
        for (int kt = 0; kt < 7; ++kt) {
            const int k0 = kt * 4 + (hi ? 2 : 0);
            v2f bf;
            bf.x = dws[lr * 28 + k0];
            bf.y = dws[lr * 28 + k0 + 1];
            #pragma unroll
            for (int t = 0; t < 2; ++t) {
                v2f af;
                af.x = bT[(t * 16 + lr) * 28 + k0];
                af.y = bT[(t * 16 + lr) * 28 + k0 + 1];
                accb[t] = wmma_f32(af, bf, accb[t]);
            }
        }
        #pragma unroll
        for (int t = 0; t < 2; ++t)
            #pragma unroll
            for (int r = 0; r < 8; ++r) {
                const int n = t * 16 + r + (hi ? 8 : 0);
                bmat[n * 16 + lr] = accb[t][r];
            }
    }
    __syncthreads();

    // ---- Phase C: dout = gather @ Bmat ------------------------------------
    const float db = disco_b[lr];
    v8f acc[4];
    {
        v8f z = {db, db, db, db, db, db, db, db};
        #pragma unroll
        for (int t = 0; t < 4; ++t) acc[t] = z;
    }
    const long xbase = (long)b * PP;
    #pragma unroll
    for (int kt = 0; kt < 7; ++kt) {
        const int n0 = kt * 4 + (hi ? 2 : 0);
        const float* pa = x + (xbase + nbrx[n0]) * MD;       // always valid
        const float* pb = x + (xbase + nbrx[n0 + 1]) * MD;   // always valid
        v2f bf;
        bf.x = bmat[n0 * 16 + lr];          // rows >= 25 are zero
        bf.y = bmat[(n0 + 1) * 16 + lr];
        v2f af[4];
        #pragma unroll
        for (int t = 0; t < 4; ++t) {
            af[t].x = pa[t * 16 + lr];
            af[t].y = pb[t * 16 + lr];
        }
        #pragma unroll
        for (int t = 0; t < 4; ++t) acc[t] = wmma_f32(af[t], bf, acc[t]);
    }

    // ---- Phase D: FiLM, stage to LDS --------------------------------------
    const int hrow = p / WW;
    const float* mrow = mod + (b * HH_ + hrow) * (2 * SHN);
    const float gm = mrow[lr];
    const float bt = mrow[SHN + lr];
    #pragma unroll
    for (int t = 0; t < 4; ++t)
        #pragma unroll
        for (int r = 0; r < 8; ++r) {
            const int c = t * 16 + r + (hi ? 8 : 0);
            fbuf[c * 17 + lr] = acc[t][r] * gm + bt;
        }
    __syncthreads();

    // ---- Phase E: h1 = gelu(blockdiag GEMM), 2 heads per 16-row M-tile ----
    #pragma unroll
    for (int t = 0; t < 4; ++t) {
        #pragma unroll
        for (int nt = 0; nt < 2; ++nt) {
            const float binit = hb1[(2 * t + (hi ? 1 : 0)) * HHID + nt * 16 + lr];
            v8f acch = {binit, binit, binit, binit, binit, binit, binit, binit};
            #pragma unroll
            for (int kt = 0; kt < 8; ++kt) {
                const int  k0   = kt * 4 + (hi ? 2 : 0);
                const int  s0   = (kt < 4) ? k0 : (k0 - 16);   // 0..15, safe
                const int  hsel = 2 * t + ((kt < 4) ? 0 : 1);
                const bool keep = (kt < 4) ? (lr < 8) : (lr >= 8);
                const float va0 = fbuf[(t * 16 + lr) * 17 + s0];
                const float va1 = fbuf[(t * 16 + lr) * 17 + s0 + 1];
                v2f af, bf;
                af.x = keep ? va0 : 0.f;
                af.y = keep ? va1 : 0.f;
                bf.x = hw1[(hsel * SHN + s0) * HHID + nt * 16 + lr];
                bf.y = hw1[(hsel * SHN + s0 + 1) * HHID + nt * 16 + lr];
                acch = wmma_f32(af, bf, acch);
            }
            #pragma unroll
            for (int r = 0; r < 8; ++r) {
                const int c = t * 16 + r + (hi ? 8 : 0);
                h1buf[c * 33 + nt * 16 + lr] = gelu_f(acch[r]);
            }
        }
    }
    __syncthreads();

    // ---- Phase F: h2 reduction + residual ---------------------------------
    for (int cc = l; cc < CC; cc += 32) {
        const int n = cc >> 3;
        float h2 = hb2[n];
        #pragma unroll
        for (int o = 0; o < HHID; ++o)
            h2 += h1buf[cc * 33 + o] * hw2[n * HHID + o];
        const float res = x[((long)b * PP + p) * MD + cc];
        xln[(long)q * CC + cc] = h2 + res;
    }
}

// ---------------------------------------------------------------------------
// Kernel 3: FFN over 16-point tiles via WMMA (branch-free padding)
// ---------------------------------------------------------------------------
__global__ __launch_bounds__(32)
void ffn_kernel(const float* __restrict__ x, const float* __restrict__ xln,
                const float* __restrict__ w1, const float* __restrict__ b1,
                const float* __restrict__ w2, const float* __restrict__ b2,
                float* __restrict__ out)
{
    __shared__ float xf[16 * 72];     // x_full tile, zero-padded cols 66..67
    __shared__ float hid[16 * 260];   // gelu'd hidden tile

    const int  l   = threadIdx.x;
    const long gq0 = (long)blockIdx.x * 16;

    for (int e = l; e < 16 * 68; e += 32) {
        const int m = e / 68, c = e - m * 68;
        const long gq = gq0 + m;
        float v;
        if (c < CC)      v = xln[gq * CC + c];
        else if (c < MD) v = x[gq * MD + c];
        else             v = 0.f;
        xf[m * 72 + c] = v;
    }
    __syncthreads();

    const bool hi = l >= 16;
    const int  lr = l & 15;

    for (int nt = 0; nt < 16; ++nt) {
        const float bb = b1[nt * 16 + lr];
        v8f acc = {bb, bb, bb, bb, bb, bb, bb, bb};
        #pragma unroll
        for (int kt = 0; kt < 17; ++kt) {
            const int k0 = kt * 4 + (hi ? 2 : 0);
            const int ka = (k0     < MD) ? k0     : 0;   // clamp -> safe load
            const int kb = (k0 + 1 < MD) ? k0 + 1 : 0;
            v2f af, bf;
            af.x = xf[lr * 72 + k0];                     // zero-padded rows
            af.y = xf[lr * 72 + k0 + 1];
            const float w0 = w1[(long)ka * FFNH + nt * 16 + lr];
            const float w1v = w1[(long)kb * FFNH + nt * 16 + lr];
            bf.x = (k0     < MD) ? w0  : 0.f;
            bf.y = (k0 + 1 < MD) ? w1v : 0.f;
            acc = wmma_f32(af, bf, acc);
        }
        #pragma unroll
        for (int r = 0; r < 8; ++r) {
            const int m = r + (hi ? 8 : 0);
            hid[m * 260 + nt * 16 + lr] = gelu_f(acc[r]);
        }
    }
    __syncthreads();

    for (int nt = 0; nt < 4; ++nt) {
        const float bb = b2[nt * 16 + lr];
        v8f acc = {bb, bb, bb, bb, bb, bb, bb, bb};
        #pragma unroll
        for (int kt = 0; kt < 64; ++kt) {
            const int k0 = kt * 4 + (hi ? 2 : 0);
            v2f af, bf;
            af.x = hid[lr * 260 + k0];
            af.y = hid[lr * 260 + k0 + 1];
            bf.x = w2[k0 * CC + nt * 16 + lr];
            bf.y = w2[(k0 + 1) * CC + nt * 16 + lr];
            acc = wmma_f32(af, bf, acc);
        }
        #pragma unroll
        for (int r = 0; r < 8; ++r) {
            const int  m  = r + (hi ? 8 : 0);
            const long gq = gq0 + m;
            const int  c  = nt * 16 + lr;
            out[gq * MD + c] = acc[r] + xln[gq * CC + c];
        }
    }
    {   // pass through sin_cos
        const int  m  = l >> 1, c2 = l & 1;
        const long gq = gq0 + m;
        out[gq * MD + CC + c2] = x[gq * MD + CC + c2];
    }
}

// ---------------------------------------------------------------------------
extern "C" void kernel_launch(void* const* d_in, const int* in_sizes, int n_in,
                              void* d_out, int out_size, void* d_ws, size_t ws_size,
                              hipStream_t stream)
{
    const float* x       = (const float*)d_in[0];
    const int*   nbr     = (const int*)  d_in[1];
    const float* basis   = (const float*)d_in[2];
    const float* disco_w = (const float*)d_in[3];
    const float* disco_b = (const float*)d_in[4];
    const float* lw1     = (const float*)d_in[5];
    const float* lb1     = (const float*)d_in[6];
    const float* lw2     = (const float*)d_in[7];
    const float* lb2     = (const float*)d_in[8];
    const float* hw1     = (const float*)d_in[9];
    const float* hb1     = (const float*)d_in[10];
    const float* hw2     = (const float*)d_in[11];
    const float* hb2     = (const float*)d_in[12];
    const float* fw1     = (const float*)d_in[13];
    const float* fb1     = (const float*)d_in[14];
    const float* fw2     = (const float*)d_in[15];
    const float* fb2     = (const float*)d_in[16];
    float* out = (float*)d_out;

    float* mod = (float*)d_ws;                                   // B*H*32 f32
    float* xln = (float*)((char*)d_ws + (size_t)(BB * HH_ * 32) * sizeof(float)); // B*P*64

    mod_kernel<<<BB * HH_, 32, 0, stream>>>(x, lw1, lb1, lw2, lb2, mod);
    disco_head_kernel<<<BB * PP, 32, 0, stream>>>(x, nbr, basis, disco_w, disco_b,
                                                  hw1, hb1, hw2, hb2, mod, xln);
    ffn_kernel<<<(BB * PP) / 16, 32, 0, stream>>>(x, xln, fw1, fb1, fw2, fb2, out);
}